// MPS_66760971649501
// MI455X (gfx1250) — compile-verified
//
#include <hip/hip_runtime.h>
#include <hip/hip_bf16.h>

typedef __attribute__((ext_vector_type(2))) float v2f;
typedef __attribute__((ext_vector_type(4))) float v4f;
typedef __attribute__((ext_vector_type(8))) float v8f;

#define MPS_D 512
#define MPS_M ((size_t)MPS_D * MPS_D)   // 262144 elements = 1 MB per matrix

// ---------------------------------------------------------------------------
// Init: cont[j][k] = sum_s left[s,0,j] * conj(left[s,0,k])   (D x D complex)
// ---------------------------------------------------------------------------
__global__ void mps_init_kernel(const float* __restrict__ lr,
                                const float* __restrict__ li,
                                float* __restrict__ Cr,
                                float* __restrict__ Ci) {
    int idx = blockIdx.x * blockDim.x + threadIdx.x;   // over D*D
    int j = idx >> 9;
    int k = idx & (MPS_D - 1);
    float cr = 0.f, ci = 0.f;
#pragma unroll
    for (int s = 0; s < 2; ++s) {
        float ajr = lr[s * MPS_D + j], aji = li[s * MPS_D + j];
        float bkr = lr[s * MPS_D + k], bki = li[s * MPS_D + k];
        // l_j * conj(l_k)
        cr += ajr * bkr + aji * bki;
        ci += aji * bkr - ajr * bki;
    }
    Cr[j * MPS_D + k] = cr;
    Ci[j * MPS_D + k] = ci;
}

// ---------------------------------------------------------------------------
// Phase 1:  T[s] = A[s]^T * C      (complex, D x D, per s)
//   T[s][k][j] = sum_i A[s][i][k] * C[i][j]
//   Tr = Ar^T Cr - Ai^T Ci ;  Ti = Ar^T Ci + Ai^T Cr
// One wave computes a 16x32 output strip (1 A-op tile reused over 2 B tiles).
// Also prefetches the NEXT site's bulk chunk into L2 (global_prefetch_b8):
// 4 MB = 32768 lines, two 128B lines (R and I) per thread with tid < 16384.
// ---------------------------------------------------------------------------
__global__ void __launch_bounds__(128)
mps_phase1_kernel(const float* __restrict__ Ar,  // (2,D,D)
                  const float* __restrict__ Ai,
                  const float* __restrict__ Cr,  // (D,D)
                  const float* __restrict__ Ci,
                  float* __restrict__ Tr,        // (2,D,D)
                  float* __restrict__ Ti,
                  const char* __restrict__ nextR,  // next site bulk_r chunk (2MB) or null
                  const char* __restrict__ nextI) {
    const int D = MPS_D;
    // --- prefetch next site's tensor while this site computes -------------
    if (nextR != nullptr) {
        int flatBlk = (blockIdx.z * gridDim.y + blockIdx.y) * gridDim.x + blockIdx.x;
        size_t tid = (size_t)flatBlk * blockDim.x + threadIdx.x;   // 0..32767
        if (tid < 16384) {
            __builtin_prefetch(nextR + tid * 128, 0, 1);
            __builtin_prefetch(nextI + tid * 128, 0, 1);
        }
    }

    int lane = threadIdx.x & 31;
    int wv   = threadIdx.x >> 5;          // 0..3
    int kt   = blockIdx.x;                // 0..31  output row tile (k)
    int jtp  = blockIdx.y * 4 + wv;       // 0..15  pair of output col tiles
    int s    = blockIdx.z;                // 0..1
    const float* ar = Ar + (size_t)s * D * D;
    const float* ai = Ai + (size_t)s * D * D;
    float* tr = Tr + (size_t)s * D * D;
    float* ti = Ti + (size_t)s * D * D;

    int kg = lane >> 4;                   // K-half selector
    int lm = lane & 15;
    int colK = kt * 16 + lm;              // column of A == output row k
    int colJ[2] = { jtp * 32 + lm, jtp * 32 + 16 + lm };

    v8f accR[2] = {v8f{}, v8f{}};
    v8f accI[2] = {v8f{}, v8f{}};
    for (int i0 = 0; i0 < D; i0 += 4) {
        int i = i0 + 2 * kg;
        const float* r0 = ar + (size_t)i * D;
        const float* q0 = ai + (size_t)i * D;
        // A-op (16x4): Aop[row=lm][K=2kg+v] = A[(i+v)*D + colK]
        v2f va_r = { r0[colK], r0[D + colK] };
        v2f va_i = { q0[colK], q0[D + colK] };
        v2f va_in = -va_i;  // f32 WMMA has no A/B NEG modifier (ISA: CNeg only)
        const float* cr0 = Cr + (size_t)i * D;
        const float* ci0 = Ci + (size_t)i * D;
#pragma unroll
        for (int jb = 0; jb < 2; ++jb) {
            // B-op (4x16): Bop[K=2kg+v][col=lm] = C[(i+v)*D + colJ]
            v2f vb_r = { cr0[colJ[jb]], cr0[D + colJ[jb]] };
            v2f vb_i = { ci0[colJ[jb]], ci0[D + colJ[jb]] };
            accR[jb] = __builtin_amdgcn_wmma_f32_16x16x4_f32(false, va_r,  false, vb_r,
                                                             (short)0, accR[jb], false, false);
            accR[jb] = __builtin_amdgcn_wmma_f32_16x16x4_f32(false, va_in, false, vb_i,
                                                             (short)0, accR[jb], false, false);
            accI[jb] = __builtin_amdgcn_wmma_f32_16x16x4_f32(false, va_r,  false, vb_i,
                                                             (short)0, accI[jb], false, false);
            accI[jb] = __builtin_amdgcn_wmma_f32_16x16x4_f32(false, va_i,  false, vb_r,
                                                             (short)0, accI[jb], false, false);
        }
    }
    // C/D layout: VGPR v holds row M = 8*kg + v, col = lm
#pragma unroll
    for (int jb = 0; jb < 2; ++jb) {
#pragma unroll
        for (int v = 0; v < 8; ++v) {
            int row = kt * 16 + kg * 8 + v;
            tr[(size_t)row * D + colJ[jb]] = accR[jb][v];
            ti[(size_t)row * D + colJ[jb]] = accI[jb][v];
        }
    }
}

// ---------------------------------------------------------------------------
// Phase 2 (split-s):  P[s] = T[s] * conj(A[s])   (complex, D x D, per s)
//   P[s]r = Tr Ar + Ti Ai ;  P[s]i = Ti Ar - Tr Ai
// s moved to blockIdx.z so the step exposes 1024 waves (latency hiding on the
// serial chain); a cheap combine kernel sums the two partials.
// ---------------------------------------------------------------------------
__global__ void __launch_bounds__(128)
mps_phase2_kernel(const float* __restrict__ Ar,  // (2,D,D)
                  const float* __restrict__ Ai,
                  const float* __restrict__ Tr,  // (2,D,D)
                  const float* __restrict__ Ti,
                  float* __restrict__ Pr,        // (2,D,D) partials
                  float* __restrict__ Pi) {
    const int D = MPS_D;
    int lane = threadIdx.x & 31;
    int wv   = threadIdx.x >> 5;
    int kt   = blockIdx.x;                // output row tile (k)
    int ltp  = blockIdx.y * 4 + wv;       // pair of output col tiles
    int s    = blockIdx.z;                // 0..1

    const float* ar = Ar + (size_t)s * D * D;
    const float* ai = Ai + (size_t)s * D * D;
    const float* tr = Tr + (size_t)s * D * D;
    const float* ti = Ti + (size_t)s * D * D;
    float* pr = Pr + (size_t)s * D * D;
    float* pi = Pi + (size_t)s * D * D;

    int kg = lane >> 4;
    int lm = lane & 15;
    int rowK = kt * 16 + lm;              // row of T == output row k
    int colL[2] = { ltp * 32 + lm, ltp * 32 + 16 + lm };

    v8f accR[2] = {v8f{}, v8f{}};
    v8f accI[2] = {v8f{}, v8f{}};
    for (int j0 = 0; j0 < D; j0 += 4) {
        int j = j0 + 2 * kg;
        // A-op rows are rows of T: contiguous 8B pair along j
        v2f vt_r = *(const v2f*)(tr + (size_t)rowK * D + j);
        v2f vt_i = *(const v2f*)(ti + (size_t)rowK * D + j);
        v2f vt_rn = -vt_r;
        const float* br0 = ar + (size_t)j * D;
        const float* bi0 = ai + (size_t)j * D;
#pragma unroll
        for (int lb = 0; lb < 2; ++lb) {
            // B-op rows are rows j of A
            v2f vb_r = { br0[colL[lb]], br0[D + colL[lb]] };
            v2f vb_i = { bi0[colL[lb]], bi0[D + colL[lb]] };
            accR[lb] = __builtin_amdgcn_wmma_f32_16x16x4_f32(false, vt_r,  false, vb_r,
                                                             (short)0, accR[lb], false, false);
            accR[lb] = __builtin_amdgcn_wmma_f32_16x16x4_f32(false, vt_i,  false, vb_i,
                                                             (short)0, accR[lb], false, false);
            accI[lb] = __builtin_amdgcn_wmma_f32_16x16x4_f32(false, vt_i,  false, vb_r,
                                                             (short)0, accI[lb], false, false);
            accI[lb] = __builtin_amdgcn_wmma_f32_16x16x4_f32(false, vt_rn, false, vb_i,
                                                             (short)0, accI[lb], false, false);
        }
    }
#pragma unroll
    for (int lb = 0; lb < 2; ++lb) {
#pragma unroll
        for (int v = 0; v < 8; ++v) {
            int row = kt * 16 + kg * 8 + v;
            pr[(size_t)row * D + colL[lb]] = accR[lb][v];
            pi[(size_t)row * D + colL[lb]] = accI[lb][v];
        }
    }
}

// ---------------------------------------------------------------------------
// Combine: C' = P[0] + P[1]   (vectorized 16B, pure bandwidth, ~6 MB)
// ---------------------------------------------------------------------------
__global__ void mps_combine_kernel(const float* __restrict__ Pr,
                                   const float* __restrict__ Pi,
                                   float* __restrict__ CrOut,
                                   float* __restrict__ CiOut) {
    size_t idx = ((size_t)blockIdx.x * blockDim.x + threadIdx.x) * 4;  // float4 slots
    v4f r0 = *(const v4f*)(Pr + idx);
    v4f r1 = *(const v4f*)(Pr + MPS_M + idx);
    v4f i0 = *(const v4f*)(Pi + idx);
    v4f i1 = *(const v4f*)(Pi + MPS_M + idx);
    *(v4f*)(CrOut + idx) = r0 + r1;
    *(v4f*)(CiOut + idx) = i0 + i1;
}

// ---------------------------------------------------------------------------
// Final: out = Re( sum_s sum_{i,j} c[i][j] * r[s,i] * conj(r[s,j]) )
// Single block, LDS tree reduction (deterministic).
// ---------------------------------------------------------------------------
__global__ void mps_final_kernel(const float* __restrict__ Cr,
                                 const float* __restrict__ Ci,
                                 const float* __restrict__ rr,  // (2,D,1)
                                 const float* __restrict__ ri,
                                 float* __restrict__ out) {
    __shared__ float red[1024];
    const int D = MPS_D;
    float acc = 0.f;
    for (int idx = threadIdx.x; idx < 2 * D * D; idx += blockDim.x) {
        int s = idx >> 18;
        int rem = idx & (D * D - 1);
        int i = rem >> 9;
        int j = rem & (D - 1);
        float cr = Cr[(size_t)i * D + j], ci = Ci[(size_t)i * D + j];
        float air = rr[s * D + i], aii = ri[s * D + i];
        float bjr = rr[s * D + j], bji = ri[s * D + j];
        // u = r_i * conj(r_j)
        float ur = air * bjr + aii * bji;
        float ui = aii * bjr - air * bji;
        acc += cr * ur - ci * ui;
    }
    red[threadIdx.x] = acc;
    __syncthreads();
    for (int off = blockDim.x >> 1; off > 0; off >>= 1) {
        if ((int)threadIdx.x < off) red[threadIdx.x] += red[threadIdx.x + off];
        __syncthreads();
    }
    if (threadIdx.x == 0) out[0] = red[0];
}

// ---------------------------------------------------------------------------
extern "C" void kernel_launch(void* const* d_in, const int* in_sizes, int n_in,
                              void* d_out, int out_size, void* d_ws, size_t ws_size,
                              hipStream_t stream) {
    (void)in_sizes; (void)n_in; (void)out_size; (void)ws_size;
    const int D = MPS_D;
    const int L = 128;
    const size_t M = MPS_M;

    const float* left_r  = (const float*)d_in[0];
    const float* left_i  = (const float*)d_in[1];
    const float* bulk_r  = (const float*)d_in[2];
    const float* bulk_i  = (const float*)d_in[3];
    const float* right_r = (const float*)d_in[4];
    const float* right_i = (const float*)d_in[5];

    // Workspace (floats): C0r C0i C1r C1i (4M) | Tr Ti (4M) | Pr Pi (4M) = 12 MB
    float* ws  = (float*)d_ws;
    float* C0r = ws + 0 * M;
    float* C0i = ws + 1 * M;
    float* C1r = ws + 2 * M;
    float* C1i = ws + 3 * M;
    float* Tr  = ws + 4 * M;   // (2,D,D)
    float* Ti  = ws + 6 * M;   // (2,D,D)
    float* Pr  = ws + 8 * M;   // (2,D,D)
    float* Pi  = ws + 10 * M;  // (2,D,D)

    mps_init_kernel<<<dim3((D * D) / 256), dim3(256), 0, stream>>>(left_r, left_i,
                                                                   C0r, C0i);

    float* inR = C0r; float* inI = C0i;
    float* outR = C1r; float* outI = C1i;
    dim3 g12(D / 16, D / (32 * 4), 2);  // (kt, tile-pair groups, s): 1024 waves
    dim3 blk(128);                      // 4 waves
    dim3 gcmb(D * D / (4 * 256));       // combine: float4 per thread

    for (int site = 0; site < L - 2; ++site) {
        const float* Ar = bulk_r + (size_t)site * 2 * M;
        const float* Ai = bulk_i + (size_t)site * 2 * M;
        const char* nR = (site + 1 < L - 2) ? (const char*)(bulk_r + (size_t)(site + 1) * 2 * M)
                                            : nullptr;
        const char* nI = (site + 1 < L - 2) ? (const char*)(bulk_i + (size_t)(site + 1) * 2 * M)
                                            : nullptr;
        mps_phase1_kernel<<<g12, blk, 0, stream>>>(Ar, Ai, inR, inI, Tr, Ti, nR, nI);
        mps_phase2_kernel<<<g12, blk, 0, stream>>>(Ar, Ai, Tr, Ti, Pr, Pi);
        mps_combine_kernel<<<gcmb, dim3(256), 0, stream>>>(Pr, Pi, outR, outI);
        float* t;
        t = inR; inR = outR; outR = t;
        t = inI; inI = outI; outI = t;
    }

    mps_final_kernel<<<dim3(1), dim3(1024), 0, stream>>>(inR, inI, right_r, right_i,
                                                         (float*)d_out);
}